// RNN_66563403153708
// MI455X (gfx1250) — compile-verified
//
#include <hip/hip_runtime.h>

typedef __attribute__((ext_vector_type(2))) float v2f;
typedef __attribute__((ext_vector_type(8))) float v8f;

namespace {
constexpr int kB = 64;
constexpr int kT = 512;
constexpr int kF = 1024;
constexpr int kU = 1024;
constexpr unsigned kScanBlocks = 64;   // 16 x 4 grid, must all co-reside
}

static __device__ __forceinline__ v8f wmma4(v2f a, v2f b, v8f c) {
    // 8 args: (neg_a, A, neg_b, B, c_mod, C, reuse_a, reuse_b)
    return __builtin_amdgcn_wmma_f32_16x16x4_f32(false, a, false, b, (short)0, c,
                                                 false, false);
}

// ---------------------------------------------------------------------------
// Phase 1: xproj = sequences[M=B*T, K=F] @ kernel[K=F, N=U], written straight
// into d_out (row-major [B,T,U] == [M,U]).
// One wave computes a 32x32 block (2x2 WMMA tiles): per K-step, 2 A-frag +
// 2 B-frag loads feed 4 WMMAs -> ~8 FLOP/B from cache, dep distance 4.
// 4 waves per block in 2x2 -> 64x64 block tile. grid = (N/64, M/64).
// ---------------------------------------------------------------------------
__global__ __launch_bounds__(128, 1)
void rnn_xproj(const float* __restrict__ A, const float* __restrict__ W,
               float* __restrict__ out) {
    const int lane = threadIdx.x & 31;
    const int wave = threadIdx.x >> 5;
    const int l15  = lane & 15;
    const int half = lane >> 4;   // 0: lanes 0-15, 1: lanes 16-31

    const int m0 = (blockIdx.y << 6) + ((wave >> 1) << 5);  // 32-row block
    const int n0 = (blockIdx.x << 6) + ((wave & 1) << 5);   // 32-col block

    // A fragment (16x4 f32): lane half selects K pair {k,k+1} vs {k+2,k+3}
    const float* ap = A + (size_t)(m0 + l15) * kF + (half << 1);
    // B fragment (4x16 f32): v0 = row kk, v1 = row kk+1, kk = k + 2*half
    const float* bp = W + (size_t)(half << 1) * kU + (n0 + l15);

    v8f c00 = {}, c01 = {}, c10 = {}, c11 = {};

    #pragma unroll 4
    for (int k = 0; k < kF; k += 4) {
        v2f a0 = *(const v2f*)(ap + k);
        v2f a1 = *(const v2f*)(ap + 16 * kF + k);
        const float* bk = bp + (size_t)k * kU;
        v2f b0, b1;
        b0.x = bk[0];       b0.y = bk[kU];
        b1.x = bk[16];      b1.y = bk[kU + 16];
        c00 = wmma4(a0, b0, c00);
        c01 = wmma4(a0, b1, c01);
        c10 = wmma4(a1, b0, c10);
        c11 = wmma4(a1, b1, c11);
    }

    // C/D layout: VGPR v -> row mBase + v + 8*half, col nBase + l15
    const int r0 = m0 + (half << 3);
    #pragma unroll
    for (int v = 0; v < 8; ++v) {
        out[(size_t)(r0 + v) * kU + n0 + l15]           = c00[v];
        out[(size_t)(r0 + v) * kU + n0 + 16 + l15]      = c01[v];
        out[(size_t)(r0 + 16 + v) * kU + n0 + l15]      = c10[v];
        out[(size_t)(r0 + 16 + v) * kU + n0 + 16 + l15] = c11[v];
    }
}

// ---------------------------------------------------------------------------
// Phase 2: persistent scan kernel. All kT steps run inside one launch:
//   out[:, t, :] += H_t @ R,   H_0 = h0, H_t = out[:, t-1, :]
// grid = (16, 4) = 64 blocks x 128 threads (trivially co-resident); one
// 16x16 tile per wave, 4 interleaved K-chains. Steps are separated by a
// grid-wide arrival barrier in d_ws (release fence -> atomicAdd -> spin with
// s_sleep -> acquire fence), which both orders execution and makes the new
// h slice visible across WGPs through L2.
// ---------------------------------------------------------------------------
__global__ __launch_bounds__(128, 1)
void rnn_scan(const float* __restrict__ h0, const float* __restrict__ R,
              float* __restrict__ out, unsigned* __restrict__ ctr) {
    const int lane = threadIdx.x & 31;
    const int wave = threadIdx.x >> 5;
    const int l15  = lane & 15;
    const int half = lane >> 4;

    const int m0 = blockIdx.y << 4;                  // batch tile
    const int n0 = ((blockIdx.x << 2) + wave) << 4;  // unit tile
    const int row0 = m0 + (half << 3);

    // R fragment base: column n0+l15, rows kk=k+2*half and kk+1
    const float* bp = R + (size_t)(half << 1) * kU + (n0 + l15);

    for (int t = 0; t < kT; ++t) {
        const float* hbase   = (t == 0) ? h0 : (out + (size_t)(t - 1) * kU);
        const size_t hstride = (t == 0) ? (size_t)kU : (size_t)kT * kU;
        const float* ap = hbase + (size_t)(m0 + l15) * hstride + (half << 1);
        float* op = out + (size_t)t * kU + n0 + l15;

        // c0 starts as the xproj value already living in out[:, t, :]
        v8f c0, c1 = {}, c2 = {}, c3 = {};
        #pragma unroll
        for (int v = 0; v < 8; ++v)
            c0[v] = op[(size_t)(row0 + v) * (kT * kU)];

        #pragma unroll 4
        for (int k = 0; k < kU; k += 16) {
            v2f a0 = *(const v2f*)(ap + k);
            v2f a1 = *(const v2f*)(ap + k + 4);
            v2f a2 = *(const v2f*)(ap + k + 8);
            v2f a3 = *(const v2f*)(ap + k + 12);
            const float* bk = bp + (size_t)k * kU;
            v2f b0, b1, b2, b3;
            b0.x = bk[0];           b0.y = bk[kU];
            b1.x = bk[4 * kU];      b1.y = bk[5 * kU];
            b2.x = bk[8 * kU];      b2.y = bk[9 * kU];
            b3.x = bk[12 * kU];     b3.y = bk[13 * kU];
            c0 = wmma4(a0, b0, c0);
            c1 = wmma4(a1, b1, c1);
            c2 = wmma4(a2, b2, c2);
            c3 = wmma4(a3, b3, c3);
        }
        v8f c = (c0 + c1) + (c2 + c3);

        #pragma unroll
        for (int v = 0; v < 8; ++v)
            op[(size_t)(row0 + v) * (kT * kU)] = c[v];

        if (t + 1 < kT) {
            // ---- grid-wide step barrier ----
            __threadfence();   // release: drain my stores to device scope
            __syncthreads();
            if (threadIdx.x == 0) {
                __hip_atomic_fetch_add(ctr + t, 1u, __ATOMIC_RELEASE,
                                       __HIP_MEMORY_SCOPE_AGENT);
                while (__hip_atomic_load(ctr + t, __ATOMIC_ACQUIRE,
                                         __HIP_MEMORY_SCOPE_AGENT) < kScanBlocks) {
                    __builtin_amdgcn_s_sleep(1);
                }
            }
            __syncthreads();
            __threadfence();   // acquire: invalidate so H loads see fresh data
        }
    }
}

extern "C" void kernel_launch(void* const* d_in, const int* in_sizes, int n_in,
                              void* d_out, int out_size, void* d_ws, size_t ws_size,
                              hipStream_t stream) {
    (void)in_sizes; (void)n_in; (void)out_size; (void)ws_size;

    const float* seq = (const float*)d_in[0];  // [B, T, F]
    const float* Wk  = (const float*)d_in[1];  // [F, U]
    const float* Rk  = (const float*)d_in[2];  // [U, U]
    const float* h0  = (const float*)d_in[3];  // [B, U]
    float* out = (float*)d_out;                // [B, T, U]
    unsigned* ctr = (unsigned*)d_ws;           // kT arrival counters

    // Zero the per-step barrier counters (graph-capture legal).
    hipMemsetAsync(d_ws, 0, kT * sizeof(unsigned), stream);

    dim3 blk(128, 1, 1);

    // Phase 1: big GEMM, result placed directly in d_out.
    rnn_xproj<<<dim3(kU / 64, (kB * kT) / 64), blk, 0, stream>>>(seq, Wk, out);

    // Phase 2: one persistent launch for the whole sequential scan.
    rnn_scan<<<dim3(kU / 64, kB / 16), blk, 0, stream>>>(h0, Rk, out, ctr);
}